// BaseTransform_84593675862568
// MI455X (gfx1250) — compile-verified
//
#include <hip/hip_runtime.h>
#include <hip/hip_bf16.h>

typedef float v2f __attribute__((ext_vector_type(2)));
typedef float v8f __attribute__((ext_vector_type(8)));

#define BSZ   2
#define NCAM  6
#define DNUM  59
#define FH    32
#define FW    88
#define CCH   80
#define NXV   360
#define NYV   360
#define PTS   (BSZ * NCAM * DNUM * FH * FW)   /* 1,993,728 */
#define NCHUNK (PTS / 16)                     /* 124,608 — exact */
#define PTS_PER_B (NCAM * DNUM * FH * FW)     /* 996,864 */

// Broadcast lane k (compile-time constant) to all lanes via v_readlane_b32.
// Result is wave-uniform (SGPR), so downstream bounds tests go scalar.
__device__ __forceinline__ float lane_bcast(float v, int srclane) {
  return __uint_as_float(__builtin_amdgcn_readlane(__float_as_uint(v), srclane));
}

// ---------------------------------------------------------------------------
// Kernel 1: zero the output accumulator (83 MB), vectorized.
// ---------------------------------------------------------------------------
__global__ void zero_out_kernel(float* __restrict__ out, int n) {
  int i4 = (blockIdx.x * blockDim.x + threadIdx.x) * 4;
  if (i4 + 3 < n) {
    float4 z; z.x = 0.f; z.y = 0.f; z.z = 0.f; z.w = 0.f;
    *reinterpret_cast<float4*>(out + i4) = z;
  } else {
    for (int j = i4; j < n; ++j) out[j] = 0.f;
  }
}

// ---------------------------------------------------------------------------
// Kernel 2: invert the 12 lidar2img 4x4 matrices (adjugate method) into ws.
// Row-major in, row-major inverse out.
// ---------------------------------------------------------------------------
__global__ void invert_l2i_kernel(const float* __restrict__ l2i,
                                  float* __restrict__ minv) {
  int i = threadIdx.x;
  if (i >= BSZ * NCAM) return;
  const float* m = l2i + i * 16;
  float inv[16];
  inv[0]  =  m[5]*m[10]*m[15] - m[5]*m[11]*m[14] - m[9]*m[6]*m[15] + m[9]*m[7]*m[14] + m[13]*m[6]*m[11] - m[13]*m[7]*m[10];
  inv[4]  = -m[4]*m[10]*m[15] + m[4]*m[11]*m[14] + m[8]*m[6]*m[15] - m[8]*m[7]*m[14] - m[12]*m[6]*m[11] + m[12]*m[7]*m[10];
  inv[8]  =  m[4]*m[9]*m[15]  - m[4]*m[11]*m[13] - m[8]*m[5]*m[15] + m[8]*m[7]*m[13] + m[12]*m[5]*m[11] - m[12]*m[7]*m[9];
  inv[12] = -m[4]*m[9]*m[14]  + m[4]*m[10]*m[13] + m[8]*m[5]*m[14] - m[8]*m[6]*m[13] - m[12]*m[5]*m[10] + m[12]*m[6]*m[9];
  inv[1]  = -m[1]*m[10]*m[15] + m[1]*m[11]*m[14] + m[9]*m[2]*m[15] - m[9]*m[3]*m[14] - m[13]*m[2]*m[11] + m[13]*m[3]*m[10];
  inv[5]  =  m[0]*m[10]*m[15] - m[0]*m[11]*m[14] - m[8]*m[2]*m[15] + m[8]*m[3]*m[14] + m[12]*m[2]*m[11] - m[12]*m[3]*m[10];
  inv[9]  = -m[0]*m[9]*m[15]  + m[0]*m[11]*m[13] + m[8]*m[1]*m[15] - m[8]*m[3]*m[13] - m[12]*m[1]*m[11] + m[12]*m[3]*m[9];
  inv[13] =  m[0]*m[9]*m[14]  - m[0]*m[10]*m[13] - m[8]*m[1]*m[14] + m[8]*m[2]*m[13] + m[12]*m[1]*m[10] - m[12]*m[2]*m[9];
  inv[2]  =  m[1]*m[6]*m[15]  - m[1]*m[7]*m[14]  - m[5]*m[2]*m[15] + m[5]*m[3]*m[14] + m[13]*m[2]*m[7]  - m[13]*m[3]*m[6];
  inv[6]  = -m[0]*m[6]*m[15]  + m[0]*m[7]*m[14]  + m[4]*m[2]*m[15] - m[4]*m[3]*m[14] - m[12]*m[2]*m[7]  + m[12]*m[3]*m[6];
  inv[10] =  m[0]*m[5]*m[15]  - m[0]*m[7]*m[13]  - m[4]*m[1]*m[15] + m[4]*m[3]*m[13] + m[12]*m[1]*m[7]  - m[12]*m[3]*m[5];
  inv[14] = -m[0]*m[5]*m[14]  + m[0]*m[6]*m[13]  + m[4]*m[1]*m[14] - m[4]*m[2]*m[13] - m[12]*m[1]*m[6]  + m[12]*m[2]*m[5];
  inv[3]  = -m[1]*m[6]*m[11]  + m[1]*m[7]*m[10]  + m[5]*m[2]*m[11] - m[5]*m[3]*m[10] - m[9]*m[2]*m[7]   + m[9]*m[3]*m[6];
  inv[7]  =  m[0]*m[6]*m[11]  - m[0]*m[7]*m[10]  - m[4]*m[2]*m[11] + m[4]*m[3]*m[10] + m[8]*m[2]*m[7]   - m[8]*m[3]*m[6];
  inv[11] = -m[0]*m[5]*m[11]  + m[0]*m[7]*m[9]   + m[4]*m[1]*m[11] - m[4]*m[3]*m[9]  - m[8]*m[1]*m[7]   + m[8]*m[3]*m[5];
  inv[15] =  m[0]*m[5]*m[10]  - m[0]*m[6]*m[9]   - m[4]*m[1]*m[10] + m[4]*m[2]*m[9]  + m[8]*m[1]*m[6]   - m[8]*m[2]*m[5];
  float det = m[0]*inv[0] + m[1]*inv[4] + m[2]*inv[8] + m[3]*inv[12];
  float id = 1.0f / det;
  for (int j = 0; j < 16; ++j) minv[i * 16 + j] = inv[j] * id;
}

// ---------------------------------------------------------------------------
// Scatter one point's 80 channels. gx/gy/gz are wave-uniform (readlane), so
// the bounds test is a scalar branch; feat loads are 32 consecutive floats
// per step (coalesced 128B), streamed non-temporally; accumulation via f32
// no-return global atomics into the L2-resident output grid.
// ---------------------------------------------------------------------------
__device__ __forceinline__ void scatter_point(float* __restrict__ out,
                                              const float* __restrict__ feats,
                                              int pt, int obase,
                                              float gx, float gy, float gz,
                                              int lane) {
  // Replicate reference lower bounds / divisions in f32 exactly.
  const float LX = (-54.0f + 0.3f * 0.5f) - 0.3f * 0.5f;
  const float LZ = (-10.0f + 20.0f * 0.5f) - 20.0f * 0.5f;
  int vx = (int)((gx - LX) / 0.3f);    // trunc-toward-zero, matches astype(int32)
  int vy = (int)((gy - LX) / 0.3f);
  int vz = (int)((gz - LZ) / 20.0f);
  if ((unsigned)vx < (unsigned)NXV && (unsigned)vy < (unsigned)NYV && vz == 0) {
    long fbase = (long)pt * CCH;
    // Uniform base address of channel 0's cell; channels stride by NXV*NYV.
    int obb = (obase * NXV + vx) * NYV + vy;
#pragma unroll
    for (int cb = 0; cb < 3; ++cb) {
      int c = lane + 32 * cb;
      if (c < CCH) {
        float v = __builtin_nontemporal_load(feats + fbase + c);  // th:NT stream
        atomicAdd(out + obb + c * (NXV * NYV), v);  // global_atomic_add_f32
      }
    }
  }
}

// ---------------------------------------------------------------------------
// Kernel 3: one wave per 16 points. Geometry = V_WMMA_F32_16X16X4_F32:
//   A (16x4) = p4 vectors  [lanes 0-15: K0,K1 | lanes 16-31: K2,K3]
//   B (4x16) = img2lidar^T [cols 0..3 meaningful, rest zero]
//   D (16x16) row r -> acc VGPR r; lanes 0-15: point r, lanes 16-31: point r+8
// ---------------------------------------------------------------------------
__global__ void __launch_bounds__(256)
bev_wmma_scatter(const float* __restrict__ feats,
                 const float* __restrict__ trans,
                 const float* __restrict__ scale,
                 const float* __restrict__ minv,
                 float* __restrict__ out) {
  const int lane = threadIdx.x & 31;
  const int wave = (blockIdx.x * blockDim.x + threadIdx.x) >> 5;
  if (wave >= NCHUNK) return;

  const int ptBase = wave * 16;
  const int m = lane & 15;

  // Decode (b, n, d, h, w) for this lane's point. Chunks of 16 never cross a
  // camera boundary (D*H*W = 166144 is divisible by 16).
  unsigned pt = (unsigned)(ptBase + m);
  unsigned w  = pt % FW;  unsigned t = pt / FW;
  unsigned h  = t % FH;   t /= FH;
  unsigned dd = t % DNUM; t /= DNUM;
  unsigned nc = t % NCAM;
  unsigned b  = t / NCAM;
  const int cam = (int)(b * NCAM + nc);

  const float tx = trans[cam * 3 + 0];
  const float ty = trans[cam * 3 + 1];
  const float tz = trans[cam * 3 + 2];
  const float sc = scale[cam];

  const float fx = (float)w * (703.0f / 87.0f);   // linspace(0,703,88)
  const float fy = (float)h * (255.0f / 31.0f);   // linspace(0,255,32)
  const float d  = (1.0f + (float)dd) + tz;       // depth bins 1..59 (+tz)
  const float x  = (fx + tx) / sc;
  const float y  = (fy + ty) / sc;

  const bool hi = lane >= 16;
  v2f A;
  A.x = hi ? d    : x * d;   // K=2 : K=0
  A.y = hi ? 1.0f : y * d;   // K=3 : K=1

  // B[k][n] = Minv[n][k] (so D[m][n] = (Minv * p4_m)[n]); columns n>=4 zero.
  const float* Mi = minv + cam * 16;
  const int ncol = lane & 15;
  float b0 = 0.f, b1 = 0.f;
  if (ncol < 4) {
    const int k = hi ? 2 : 0;
    b0 = Mi[ncol * 4 + k];
    b1 = Mi[ncol * 4 + k + 1];
  }
  v2f Bm; Bm.x = b0; Bm.y = b1;

  v8f Cz = {};
  // (neg_a, A, neg_b, B, c_mod, C, reuse_a, reuse_b)
  v8f Dm = __builtin_amdgcn_wmma_f32_16x16x4_f32(
      false, A, false, Bm, (short)0, Cz, false, false);

  // Batch index / output channel base are uniform across the whole chunk.
  const int bU = (unsigned)ptBase / (unsigned)PTS_PER_B;
  const int obase = bU * CCH;

#pragma unroll
  for (int r = 0; r < 8; ++r) {
    // Point r: components live in lanes 0..2 of acc row r.
    float gx0 = lane_bcast(Dm[r], 0);
    float gy0 = lane_bcast(Dm[r], 1);
    float gz0 = lane_bcast(Dm[r], 2);
    scatter_point(out, feats, ptBase + r, obase, gx0, gy0, gz0, lane);
    // Point r+8: components live in lanes 16..18 of acc row r.
    float gx1 = lane_bcast(Dm[r], 16);
    float gy1 = lane_bcast(Dm[r], 17);
    float gz1 = lane_bcast(Dm[r], 18);
    scatter_point(out, feats, ptBase + 8 + r, obase, gx1, gy1, gz1, lane);
  }
}

// ---------------------------------------------------------------------------
extern "C" void kernel_launch(void* const* d_in, const int* in_sizes, int n_in,
                              void* d_out, int out_size, void* d_ws, size_t ws_size,
                              hipStream_t stream) {
  const float* feats = (const float*)d_in[0];  // (2,6,59,32,88,80) f32
  const float* trans = (const float*)d_in[1];  // (2,6,3)
  const float* scale = (const float*)d_in[2];  // (2,6)
  const float* l2i   = (const float*)d_in[3];  // (2,6,4,4)
  float* out  = (float*)d_out;                 // (2,80,360,360) f32
  float* minv = (float*)d_ws;                  // 12 * 16 floats scratch

  // 1) zero the accumulator grid
  int n  = out_size;
  int n4 = (n + 3) / 4;
  zero_out_kernel<<<(n4 + 255) / 256, 256, 0, stream>>>(out, n);

  // 2) invert the 12 camera matrices
  invert_l2i_kernel<<<1, 32, 0, stream>>>(l2i, minv);

  // 3) WMMA geometry + streaming scatter-add (one wave per 16 points)
  const int blocks = (NCHUNK * 32) / 256;      // 15,576 — exact cover
  bev_wmma_scatter<<<blocks, 256, 0, stream>>>(feats, trans, scale, minv, out);
}